// ContrastivePhasedGNN_8718783610907
// MI455X (gfx1250) — compile-verified
//
#include <hip/hip_runtime.h>
#include <hip/hip_bf16.h>

typedef __attribute__((ext_vector_type(2))) float v2f;
typedef __attribute__((ext_vector_type(8))) float v8f;

#define F_IN 128
#define HID  256
#define EMB  128
#define NB   1024
#define NG   8
#define NFAM 100

// ---------------- degree / norm ----------------
__global__ void k_init_deg(float* __restrict__ deg, int n) {
    int i = blockIdx.x * blockDim.x + threadIdx.x;
    if (i < n) deg[i] = 1.0f;                 // self-loop contribution
}

__global__ void k_deg_edges(const int* __restrict__ dst, float* __restrict__ deg, int E) {
    int e = blockIdx.x * blockDim.x + threadIdx.x;
    if (e < E) atomicAdd(deg + dst[e], 1.0f);
}

__global__ void k_dinv(float* __restrict__ deg, int n) {
    int i = blockIdx.x * blockDim.x + threadIdx.x;
    if (i < n) deg[i] = rsqrtf(deg[i]);       // deg >= 1 always (self loop)
}

// ---------------- fp32 WMMA GEMM: C = A[MxK] @ B[KxN] ----------------
// One wave computes a 16x64 strip of C (4 accumulators) so each A fragment
// feeds 4 V_WMMA_F32_16X16X4_F32 ops per K-step (4x WMMA density, 4x less A traffic).
// A frag (16x4): lanes 0-15 rows M=0..15 K={k0,k0+1}; lanes 16-31 K={k0+2,k0+3}.
// B frag (4x16): lanes 0-15 cols N=0..15 rows K={k0,k0+1}; lanes 16-31 rows K={k0+2,k0+3}.
// C/D (16x16 f32, 8 VGPRs): lane<16 -> M=r, N=lane; lane>=16 -> M=r+8, N=lane-16.
// Requires Nn % 64 == 0 (holds: 256 and 128).
__global__ void k_gemm_f32_wmma(const float* __restrict__ A,
                                const float* __restrict__ B,
                                float* __restrict__ C,
                                int M, int Nn, int K) {
    int wave  = blockIdx.x * (blockDim.x >> 5) + (threadIdx.x >> 5);
    int lane  = threadIdx.x & 31;
    int nblk  = Nn >> 6;                      // 64-wide column blocks
    int tm    = wave / nblk;
    int tnb   = wave - tm * nblk;
    if (tm * 16 >= M) return;                 // wave-uniform exit: EXEC stays all-1 for WMMA

    int l  = lane & 15;
    int hi = lane >> 4;                       // 0 or 1 (K-half selector)

    v8f acc0 = {}, acc1 = {}, acc2 = {}, acc3 = {};
    const float* arow = A + (size_t)(tm * 16 + l) * K + hi * 2;
    const float* bcol = B + (size_t)(hi * 2) * Nn + tnb * 64 + l;

    for (int k0 = 0; k0 < K; k0 += 4) {
        v2f a;
        a.x = arow[0];
        a.y = arow[1];
        v2f b0, b1, b2, b3;
        b0.x = bcol[0];        b0.y = bcol[Nn];
        b1.x = bcol[16];       b1.y = bcol[Nn + 16];
        b2.x = bcol[32];       b2.y = bcol[Nn + 32];
        b3.x = bcol[48];       b3.y = bcol[Nn + 48];
        acc0 = __builtin_amdgcn_wmma_f32_16x16x4_f32(false, a, false, b0, (short)0, acc0, false, false);
        acc1 = __builtin_amdgcn_wmma_f32_16x16x4_f32(false, a, false, b1, (short)0, acc1, false, false);
        acc2 = __builtin_amdgcn_wmma_f32_16x16x4_f32(false, a, false, b2, (short)0, acc2, false, false);
        acc3 = __builtin_amdgcn_wmma_f32_16x16x4_f32(false, a, false, b3, (short)0, acc3, false, false);
        arow += 4;
        bcol += (size_t)4 * Nn;
    }

    int mbase = tm * 16 + hi * 8;
    float* crow = C + (size_t)mbase * Nn + tnb * 64 + l;
#pragma unroll
    for (int r = 0; r < 8; ++r) {
        crow[(size_t)r * Nn +  0] = acc0[r];
        crow[(size_t)r * Nn + 16] = acc1[r];
        crow[(size_t)r * Nn + 32] = acc2[r];
        crow[(size_t)r * Nn + 48] = acc3[r];
    }
}

// ---------------- self-loop term (also initializes agg) ----------------
__global__ void k_self(float* __restrict__ agg, const float* __restrict__ hw,
                       const float* __restrict__ dinv, int n, int cshift) {
    size_t i = blockIdx.x * (size_t)blockDim.x + threadIdx.x;
    if (i >= ((size_t)n << cshift)) return;
    int node = (int)(i >> cshift);
    float di = dinv[node];
    agg[i] = hw[i] * di * di;
}

// ---------------- edge scatter: agg[dst] += hw[src] * dinv[src]*dinv[dst] ----------------
// Wave per edge; float4 (b128) gathers from L2-resident hw, scalar f32 atomics scatter.
__global__ void k_edge_agg(const float* __restrict__ hw, float* __restrict__ agg,
                           const int* __restrict__ src, const int* __restrict__ dst,
                           const float* __restrict__ dinv, int E, int C) {
    int e = blockIdx.x * (blockDim.x >> 5) + (threadIdx.x >> 5);
    if (e >= E) return;
    int lane = threadIdx.x & 31;
    int s = src[e], d = dst[e];
    float norm = dinv[s] * dinv[d];
    const float4* hs4 = (const float4*)(hw + (size_t)s * C);
    float* ad = agg + (size_t)d * C;
    int nvec = C >> 2;                                 // 64 (C=256) or 32 (C=128)
    for (int v = lane; v < nvec; v += 32) {
        float4 hv = hs4[v];
        int c = v << 2;
        atomicAdd(ad + c + 0, hv.x * norm);
        atomicAdd(ad + c + 1, hv.y * norm);
        atomicAdd(ad + c + 2, hv.z * norm);
        atomicAdd(ad + c + 3, hv.w * norm);
    }
}

// ---------------- bias (+optional relu), in place ----------------
__global__ void k_bias_act(float* __restrict__ h, const float* __restrict__ b,
                           int n, int C, int relu) {
    size_t i = blockIdx.x * (size_t)blockDim.x + threadIdx.x;
    if (i >= (size_t)n * C) return;
    float v = h[i] + b[i & (C - 1)];          // C is a power of two (256 or 128)
    h[i] = relu ? fmaxf(v, 0.0f) : v;
}

// ---------------- pooling ----------------
__global__ void k_zero(float* __restrict__ p, size_t n) {
    size_t i = blockIdx.x * (size_t)blockDim.x + threadIdx.x;
    if (i < n) p[i] = 0.0f;
}

__global__ void k_counts(const int* __restrict__ batch, float* __restrict__ counts, int n) {
    int i = blockIdx.x * blockDim.x + threadIdx.x;
    if (i < n) atomicAdd(counts + batch[i], 1.0f);
}

__global__ void k_pool(const float* __restrict__ h2, const int* __restrict__ batch,
                       float* __restrict__ emb_sum, int n) {
    size_t i = blockIdx.x * (size_t)blockDim.x + threadIdx.x;
    if (i >= ((size_t)n << 7)) return;
    int node = (int)(i >> 7);
    int c = (int)(i & 127);
    atomicAdd(emb_sum + (((size_t)batch[node]) << 7) + c, h2[i]);
}

__global__ void k_emb(const float* __restrict__ emb_sum, const float* __restrict__ counts,
                      float* __restrict__ emb) {
    int i = blockIdx.x * blockDim.x + threadIdx.x;
    if (i >= NB * EMB) return;
    int b = i >> 7;
    emb[i] = emb_sum[i] / fmaxf(counts[b], 1.0f);
}

// ---------------- group head + argmax ----------------
__global__ void k_group(const float* __restrict__ emb, const float* __restrict__ Wg,
                        const float* __restrict__ bg, float* __restrict__ glogits,
                        int* __restrict__ groups) {
    int b = blockIdx.x * blockDim.x + threadIdx.x;
    if (b >= NB) return;
    const float* e = emb + (size_t)b * EMB;
    float best = -__builtin_huge_valf();
    int bi = 0;
    for (int g = 0; g < NG; ++g) {
        float acc = bg[g];
        for (int k = 0; k < EMB; ++k) acc += e[k] * Wg[k * NG + g];
        glogits[b * NG + g] = acc;
        if (acc > best) { best = acc; bi = g; }   // strict '>' keeps first max (argmax semantics)
    }
    groups[b] = bi;
}

// ---------------- masked per-group family logits ----------------
__global__ void k_fam(const float* __restrict__ emb, const float* __restrict__ Wf,
                      const float* __restrict__ bf, const int* __restrict__ groups,
                      float* __restrict__ out) {
    size_t i = blockIdx.x * (size_t)blockDim.x + threadIdx.x;
    if (i >= (size_t)NG * NB * NFAM) return;
    int f = (int)(i % NFAM);
    int b = (int)((i / NFAM) % NB);
    int g = (int)(i / ((size_t)NFAM * NB));
    if (groups[b] != g) { out[i] = 0.0f; return; }
    const float* e = emb + (size_t)b * EMB;
    const float* w = Wf + (size_t)g * EMB * NFAM + f;
    float acc = bf[g * NFAM + f];
    for (int k = 0; k < EMB; ++k) acc += e[k] * w[(size_t)k * NFAM];
    out[i] = acc;
}

// ---------------- launch ----------------
extern "C" void kernel_launch(void* const* d_in, const int* in_sizes, int n_in,
                              void* d_out, int out_size, void* d_ws, size_t ws_size,
                              hipStream_t stream) {
    const float* x  = (const float*)d_in[0];
    const float* W1 = (const float*)d_in[1];
    const float* b1 = (const float*)d_in[2];
    const float* W2 = (const float*)d_in[3];
    const float* b2 = (const float*)d_in[4];
    const float* Wg = (const float*)d_in[5];
    const float* bg = (const float*)d_in[6];
    const float* Wf = (const float*)d_in[7];
    const float* bf = (const float*)d_in[8];
    const int*   ei = (const int*)d_in[9];
    const int*   batch = (const int*)d_in[10];

    const int N = in_sizes[0] / F_IN;
    const int E = in_sizes[9] / 2;
    const int* src = ei;
    const int* dst = ei + E;

    // workspace layout (floats)
    float* dinv    = (float*)d_ws;                 // N
    float* buf1    = dinv + N;                     // N*HID : hw1, later hw2 [0:N*EMB) + agg2 [N*EMB:N*HID)
    float* agg1    = buf1 + (size_t)N * HID;       // N*HID : agg1 -> h1 in place
    float* emb_sum = agg1 + (size_t)N * HID;       // NB*EMB
    float* counts  = emb_sum + NB * EMB;           // NB
    int*   groups  = (int*)(counts + NB);          // NB

    float* hw1  = buf1;
    float* hw2  = buf1;                            // reuse (only N*EMB of it)
    float* agg2 = buf1 + (size_t)N * EMB;

    float* out_emb  = (float*)d_out;               // NB*EMB
    float* out_glog = out_emb + NB * EMB;          // NB*NG
    float* out_fam  = out_glog + NB * NG;          // NG*NB*NFAM

    const int T = 256;

    // 1) degrees + normalization
    k_init_deg<<<(N + T - 1) / T, T, 0, stream>>>(dinv, N);
    k_deg_edges<<<(E + T - 1) / T, T, 0, stream>>>(dst, dinv, E);
    k_dinv<<<(N + T - 1) / T, T, 0, stream>>>(dinv, N);

    // 2) layer 1: hw1 = x @ W1 (WMMA fp32), aggregate, bias+relu
    {
        int strips = (N / 16) * (HID / 64);        // 16x64 strip per wave
        k_gemm_f32_wmma<<<(strips + 7) / 8, 256, 0, stream>>>(x, W1, hw1, N, HID, F_IN);
        size_t nc = (size_t)N * HID;
        k_self<<<(unsigned)((nc + T - 1) / T), T, 0, stream>>>(agg1, hw1, dinv, N, 8);
        k_edge_agg<<<(E + 7) / 8, 256, 0, stream>>>(hw1, agg1, src, dst, dinv, E, HID);
        k_bias_act<<<(unsigned)((nc + T - 1) / T), T, 0, stream>>>(agg1, b1, N, HID, 1);
    }
    float* h1 = agg1;

    // 3) layer 2: hw2 = h1 @ W2 (WMMA fp32), aggregate, bias
    {
        int strips = (N / 16) * (EMB / 64);
        k_gemm_f32_wmma<<<(strips + 7) / 8, 256, 0, stream>>>(h1, W2, hw2, N, EMB, HID);
        size_t nc = (size_t)N * EMB;
        k_self<<<(unsigned)((nc + T - 1) / T), T, 0, stream>>>(agg2, hw2, dinv, N, 7);
        k_edge_agg<<<(E + 7) / 8, 256, 0, stream>>>(hw2, agg2, src, dst, dinv, E, EMB);
        k_bias_act<<<(unsigned)((nc + T - 1) / T), T, 0, stream>>>(agg2, b2, N, EMB, 0);
    }
    float* h2 = agg2;

    // 4) global mean pool -> emb (written to d_out)
    k_zero<<<(NB * EMB + NB + T - 1) / T, T, 0, stream>>>(emb_sum, NB * EMB + NB);
    k_counts<<<(N + T - 1) / T, T, 0, stream>>>(batch, counts, N);
    {
        size_t nc = (size_t)N * EMB;
        k_pool<<<(unsigned)((nc + T - 1) / T), T, 0, stream>>>(h2, batch, emb_sum, N);
    }
    k_emb<<<(NB * EMB + T - 1) / T, T, 0, stream>>>(emb_sum, counts, out_emb);

    // 5) group head + argmax
    k_group<<<(NB + T - 1) / T, T, 0, stream>>>(out_emb, Wg, bg, out_glog, groups);

    // 6) masked per-group family logits
    {
        size_t tot = (size_t)NG * NB * NFAM;
        k_fam<<<(unsigned)((tot + T - 1) / T), T, 0, stream>>>(out_emb, Wf, bf, groups, out_fam);
    }
}